// GATNet_22694607192490
// MI455X (gfx1250) — compile-verified
//
#include <hip/hip_runtime.h>
#include <math.h>

#define NEG_SLOPE 0.2f
#define EPSV 1e-16f

typedef float v2f __attribute__((ext_vector_type(2)));
typedef float v8f __attribute__((ext_vector_type(8)));

// gfx1250 async global->LDS (ASYNCcnt-tracked) if the toolchain declares it
#if defined(__HIP_DEVICE_COMPILE__) && defined(__gfx1250__) && \
    __has_builtin(__builtin_amdgcn_global_load_async_to_lds_b128)
#define HAVE_ASYNC_LDS 1
// exact pointee type from the compiler diagnostic: vector_size(16) int
typedef int v4i_b128 __attribute__((vector_size(16)));
typedef __attribute__((address_space(1))) v4i_b128* gas_b128;
typedef __attribute__((address_space(3))) v4i_b128* las_b128;
#else
#define HAVE_ASYNC_LDS 0
#endif

// ---- monotonic int encoding for float atomic max (lowers to global_atomic_max_i32) ----
__device__ __forceinline__ int enc_f(float f) {
    int b = __float_as_int(f);
    return b >= 0 ? b : (b ^ 0x7FFFFFFF);
}
__device__ __forceinline__ float dec_f(int k) {
    int b = k >= 0 ? k : (k ^ 0x7FFFFFFF);
    return __int_as_float(b);
}
__device__ __forceinline__ float lrelu(float v) { return v > 0.f ? v : NEG_SLOPE * v; }

// =============================================================================
// Pack B[K x (16*NCT)] into fragment-major layout so each lane's two B values
// per k-step are adjacent: Bp[((ct*(K/4)+kb)*32 + lane)*2 + j]
//   = B[(4*kb + (lane>>4)*2 + j) * (16*NCT) + ct*16 + (lane&15)]
// Inner GEMM loop then does one coalesced 8-byte load per step.
// =============================================================================
__global__ void k_packB(const float* __restrict__ B, float* __restrict__ Bp,
                        int K, int NCT) {
    int t = blockIdx.x * blockDim.x + threadIdx.x;
    int tot = NCT * (K / 4) * 64;
    if (t >= tot) return;
    int j    = t & 1;
    int lane = (t >> 1) & 31;
    int rest = t >> 6;
    int kb   = rest % (K / 4);
    int ct   = rest / (K / 4);
    int kk   = (lane >> 4) * 2 + j;
    int col  = ct * 16 + (lane & 15);
    Bp[t] = B[(4 * kb + kk) * (16 * NCT) + col];
}

// =============================================================================
// WMMA GEMM: C[M x (16*NCT)] = A[M x K] * B (packed), fp32 via
// V_WMMA_F32_16X16X4_F32.  One block = 16-row strip; wave y = one 16-col tile.
// A tile staged in LDS (async global->LDS when available) with +4 row padding.
// =============================================================================
template <int K, int NCT>
__global__ __launch_bounds__(32 * NCT)
void gemm_wmma(const float* __restrict__ A, const float* __restrict__ Bp,
               float* __restrict__ C, int M) {
    constexpr int LDB   = 16 * NCT;
    constexpr int LDA_S = K + 4;          // pad: row pitch stays 16B-aligned, banks spread
    __shared__ float xs[16 * LDA_S];

    const int m0   = blockIdx.x * 16;
    const int tid  = threadIdx.y * 32 + threadIdx.x;
    const int nthr = 32 * NCT;

    // ---- stage 16 x K A-tile into LDS ----
#if HAVE_ASYNC_LDS
    for (int i = tid; i < (16 * K) / 4; i += nthr) {
        int r = (i * 4) / K;
        int c = (i * 4) % K;
        int gr = m0 + r; if (gr >= M) gr = M - 1;
        __builtin_amdgcn_global_load_async_to_lds_b128(
            (gas_b128)(A + (long)gr * K + c),
            (las_b128)(&xs[r * LDA_S + c]),
            0, 0);
    }
#if __has_builtin(__builtin_amdgcn_s_wait_asynccnt)
    __builtin_amdgcn_s_wait_asynccnt(0);
#else
    asm volatile("s_wait_asynccnt 0" ::: "memory");
#endif
#else
    for (int i = tid; i < (16 * K) / 4; i += nthr) {
        int r = (i * 4) / K;
        int c = (i * 4) % K;
        int gr = m0 + r; if (gr >= M) gr = M - 1;
        float4 v = *(const float4*)(A + (long)gr * K + c);
        *(float4*)(&xs[r * LDA_S + c]) = v;
    }
#endif
    __syncthreads();

    const int lane = threadIdx.x;          // wave32
    const int n0   = threadIdx.y * 16;
    const int row  = lane & 15;
    const int kk   = (lane >> 4) * 2;      // lanes 0-15: K 0,1 ; lanes 16-31: K 2,3

    // packed-B stream for this wave's col tile: 64 floats (256B/wave) per k-step
    const float* bs = Bp + ((long)threadIdx.y * (K / 4)) * 64 + lane * 2;

    v8f acc = {0.f, 0.f, 0.f, 0.f, 0.f, 0.f, 0.f, 0.f};
    #pragma unroll 4
    for (int k0 = 0; k0 < K; k0 += 4) {
        v2f a;
        a.x = xs[row * LDA_S + k0 + kk];
        a.y = xs[row * LDA_S + k0 + kk + 1];
        v2f b = *(const v2f*)bs;
        bs += 64;
        acc = __builtin_amdgcn_wmma_f32_16x16x4_f32(
            /*neg_a=*/false, a, /*neg_b=*/false, b,
            /*c_mod=*/(short)0, acc, /*reuse_a=*/false, /*reuse_b=*/false);
    }

    // C/D layout: VGPR r -> M = r (+8 for upper half-wave), N = lane&15
    const int rbase0 = (lane >> 4) ? 8 : 0;
    const int col    = n0 + row;
    if (m0 + 16 <= M) {                    // uniform fast path (branch-free stores)
        float* cp = C + (long)(m0 + rbase0) * LDB + col;
        #pragma unroll
        for (int r = 0; r < 8; ++r) cp[(long)r * LDB] = acc[r];
    } else {
        #pragma unroll
        for (int r = 0; r < 8; ++r) {
            int gr = m0 + rbase0 + r;
            if (gr < M) C[(long)gr * LDB + col] = acc[r];
        }
    }
}

// =============================================================================
// Init kernels
// =============================================================================
__global__ void k_init1(float* __restrict__ agg1, float* __restrict__ denom1,
                        int* __restrict__ m1, int Nn) {
    long t = (long)blockIdx.x * blockDim.x + threadIdx.x;
    if (t < (long)Nn * 64) agg1[t] = 0.f;
    if (t < (long)Nn * 8) { denom1[t] = 0.f; m1[t] = enc_f(-INFINITY); }
}

__global__ void k_init2(float* __restrict__ out, float* __restrict__ denom2,
                        int* __restrict__ m2, int Nn) {
    long t = (long)blockIdx.x * blockDim.x + threadIdx.x;
    if (t < (long)Nn * 40) out[t] = 0.f;
    if (t < Nn) { denom2[t] = 0.f; m2[t] = enc_f(-INFINITY); }
}

__global__ void k_padW2(const float* __restrict__ W2, float* __restrict__ W2p) {
    int t = blockIdx.x * blockDim.x + threadIdx.x;   // 64*48
    if (t >= 64 * 48) return;
    int k = t / 48, n = t % 48;
    W2p[t] = (n < 40) ? W2[k * 40 + n] : 0.f;
}

// =============================================================================
// Attention scalar kernels
// =============================================================================
__global__ void k_attn1(const float* __restrict__ h1, const float* __restrict__ as1,
                        const float* __restrict__ ad1, float* __restrict__ a_s,
                        float* __restrict__ a_d, int Nn) {
    int t = blockIdx.x * blockDim.x + threadIdx.x;   // N*8 threads: (node, head)
    if (t >= Nn * 8) return;
    int head = t & 7;
    const float* hp = h1 + (long)(t >> 3) * 64 + head * 8;
    const float* sp = as1 + head * 8;
    const float* dp = ad1 + head * 8;
    float ss = 0.f, dd = 0.f;
    #pragma unroll
    for (int c = 0; c < 8; ++c) { ss += hp[c] * sp[c]; dd += hp[c] * dp[c]; }
    a_s[t] = ss; a_d[t] = dd;
}

__global__ void k_attn2(const float* __restrict__ g2, const float* __restrict__ as2,
                        const float* __restrict__ ad2, float* __restrict__ a_s,
                        float* __restrict__ a_d, int Nn) {
    int n = blockIdx.x * blockDim.x + threadIdx.x;
    if (n >= Nn) return;
    const float* gp = g2 + (long)n * 48;
    float ss = 0.f, dd = 0.f;
    #pragma unroll
    for (int c = 0; c < 40; ++c) { ss += gp[c] * as2[c]; dd += gp[c] * ad2[c]; }
    a_s[n] = ss; a_d[n] = dd;
}

// =============================================================================
// Edge kernels (E real edges + Nn self loops).  All per-node arrays are
// L2-resident (192 MB) so gathers/atomics stay on-chip.
// =============================================================================
template <int H>
__global__ void k_edge_max(const int* __restrict__ ei, int E, int Nn,
                           const float* __restrict__ a_s, const float* __restrict__ a_d,
                           int* __restrict__ m) {
    long t = (long)blockIdx.x * blockDim.x + threadIdx.x;
    if (t >= (long)(E + Nn) * H) return;
    long e = t / H; int h = (int)(t % H);
    int src, dst;
    if (e < E) { src = ei[e]; dst = ei[E + e]; } else { src = dst = (int)(e - E); }
    float v = lrelu(a_s[(long)src * H + h] + a_d[(long)dst * H + h]);
    atomicMax(&m[(long)dst * H + h], enc_f(v));
}

template <int H>
__global__ void k_edge_sum(const int* __restrict__ ei, int E, int Nn,
                           const float* __restrict__ a_s, const float* __restrict__ a_d,
                           const int* __restrict__ m, float* __restrict__ denom) {
    long t = (long)blockIdx.x * blockDim.x + threadIdx.x;
    if (t >= (long)(E + Nn) * H) return;
    long e = t / H; int h = (int)(t % H);
    int src, dst;
    if (e < E) { src = ei[e]; dst = ei[E + e]; } else { src = dst = (int)(e - E); }
    float v = lrelu(a_s[(long)src * H + h] + a_d[(long)dst * H + h]);
    float ex = expf(v - dec_f(m[(long)dst * H + h]));
    atomicAdd(&denom[(long)dst * H + h], ex);
}

template <int H, int C>
__global__ void k_edge_agg(const int* __restrict__ ei, int E, int Nn,
                           const float* __restrict__ a_s, const float* __restrict__ a_d,
                           const int* __restrict__ m, const float* __restrict__ denom,
                           const float* __restrict__ feat, int ldf,
                           float* __restrict__ out, int ldo) {
    long t = (long)blockIdx.x * blockDim.x + threadIdx.x;
    constexpr int HC = H * C;
    if (t >= (long)(E + Nn) * HC) return;
    long e = t / HC; int hc = (int)(t % HC);
    int h = hc / C;
    int src, dst;
    if (e < E) { src = ei[e]; dst = ei[E + e]; } else { src = dst = (int)(e - E); }
    float v  = lrelu(a_s[(long)src * H + h] + a_d[(long)dst * H + h]);
    float al = expf(v - dec_f(m[(long)dst * H + h])) / (denom[(long)dst * H + h] + EPSV);
    atomicAdd(&out[(long)dst * ldo + hc], feat[(long)src * ldf + hc] * al);
}

// =============================================================================
// Pointwise epilogues
// =============================================================================
__global__ void k_bias_elu(float* __restrict__ h2, const float* __restrict__ b1, long tot) {
    long t = (long)blockIdx.x * blockDim.x + threadIdx.x;
    if (t >= tot) return;
    float v = h2[t] + b1[t & 63];
    h2[t] = v > 0.f ? v : (expf(v) - 1.f);
}

__global__ void k_logsoftmax(float* __restrict__ out, const float* __restrict__ b2, int Nn) {
    int n = blockIdx.x * blockDim.x + threadIdx.x;
    if (n >= Nn) return;
    float* row = out + (long)n * 40;
    float vals[40];
    float vmax = -INFINITY;
    #pragma unroll
    for (int c = 0; c < 40; ++c) {
        float v = row[c] + b2[c];
        vals[c] = v;
        vmax = v > vmax ? v : vmax;
    }
    float s = 0.f;
    #pragma unroll
    for (int c = 0; c < 40; ++c) s += expf(vals[c] - vmax);
    float lse = vmax + logf(s);
    #pragma unroll
    for (int c = 0; c < 40; ++c) row[c] = vals[c] - lse;
}

// =============================================================================
// Launch
// =============================================================================
static inline int cdiv_l(long a, long b) { return (int)((a + b - 1) / b); }

extern "C" void kernel_launch(void* const* d_in, const int* in_sizes, int n_in,
                              void* d_out, int out_size, void* d_ws, size_t ws_size,
                              hipStream_t stream) {
    const float* x    = (const float*)d_in[0];
    const int*   ei   = (const int*)d_in[1];
    const float* W1   = (const float*)d_in[2];
    const float* as1  = (const float*)d_in[3];
    const float* ad1  = (const float*)d_in[4];
    const float* b1   = (const float*)d_in[5];
    const float* W2   = (const float*)d_in[6];
    const float* as2  = (const float*)d_in[7];
    const float* ad2  = (const float*)d_in[8];
    const float* b2   = (const float*)d_in[9];
    float*       out  = (float*)d_out;

    const int N = in_sizes[0] / 256;
    const int E = in_sizes[1] / 2;
    const long E2 = (long)E + N;

    // workspace layout (floats)
    float* h1   = (float*)d_ws;                  // N*64
    float* aS1  = h1  + (size_t)N * 64;          // N*8
    float* aD1  = aS1 + (size_t)N * 8;           // N*8
    int*   m1   = (int*)(aD1 + (size_t)N * 8);   // N*8
    float* den1 = (float*)(m1 + (size_t)N * 8);  // N*8
    float* h2   = den1 + (size_t)N * 8;          // N*64 (agg1, then ELU in place)
    float* g2   = h2  + (size_t)N * 64;          // N*48
    float* aS2  = g2  + (size_t)N * 48;          // N
    float* aD2  = aS2 + N;                       // N
    int*   m2   = (int*)(aD2 + N);               // N
    float* den2 = (float*)(m2 + N);              // N
    float* W2p  = den2 + N;                      // 64*48
    float* W1p  = W2p + 64 * 48;                 // 16384 (packed W1)
    float* W2pp = W1p + 4 * 64 * 64;             // 3072  (packed W2p)

    const int B = 256;

    // ---------------- layer 1 ----------------
    k_init1<<<cdiv_l((long)N * 64, B), B, 0, stream>>>(h2, den1, m1, N);
    k_packB<<<cdiv_l(4 * 64 * 64, B), B, 0, stream>>>(W1, W1p, 256, 4);
    gemm_wmma<256, 4><<<cdiv_l(N, 16), dim3(32, 4), 0, stream>>>(x, W1p, h1, N);
    k_attn1<<<cdiv_l((long)N * 8, B), B, 0, stream>>>(h1, as1, ad1, aS1, aD1, N);
    k_edge_max<8><<<cdiv_l(E2 * 8, B), B, 0, stream>>>(ei, E, N, aS1, aD1, m1);
    k_edge_sum<8><<<cdiv_l(E2 * 8, B), B, 0, stream>>>(ei, E, N, aS1, aD1, m1, den1);
    k_edge_agg<8, 8><<<cdiv_l(E2 * 64, B), B, 0, stream>>>(ei, E, N, aS1, aD1, m1, den1,
                                                           h1, 64, h2, 64);
    k_bias_elu<<<cdiv_l((long)N * 64, B), B, 0, stream>>>(h2, b1, (long)N * 64);

    // ---------------- layer 2 ----------------
    k_padW2<<<cdiv_l(64 * 48, B), B, 0, stream>>>(W2, W2p);
    k_packB<<<cdiv_l(3 * 16 * 64, B), B, 0, stream>>>(W2p, W2pp, 64, 3);
    k_init2<<<cdiv_l((long)N * 40, B), B, 0, stream>>>(out, den2, m2, N);
    gemm_wmma<64, 3><<<cdiv_l(N, 16), dim3(32, 3), 0, stream>>>(h2, W2pp, g2, N);
    k_attn2<<<cdiv_l(N, B), B, 0, stream>>>(g2, as2, ad2, aS2, aD2, N);
    k_edge_max<1><<<cdiv_l(E2, B), B, 0, stream>>>(ei, E, N, aS2, aD2, m2);
    k_edge_sum<1><<<cdiv_l(E2, B), B, 0, stream>>>(ei, E, N, aS2, aD2, m2, den2);
    k_edge_agg<1, 40><<<cdiv_l(E2 * 40, B), B, 0, stream>>>(ei, E, N, aS2, aD2, m2, den2,
                                                            g2, 48, out, 40);
    k_logsoftmax<<<cdiv_l(N, B), B, 0, stream>>>(out, b2, N);
}